// CategorySpecificLinear_25795573580024
// MI455X (gfx1250) — compile-verified
//
#include <hip/hip_runtime.h>

typedef __attribute__((ext_vector_type(2))) float v2f;
typedef __attribute__((ext_vector_type(8))) float v8f;
typedef unsigned int u32x4 __attribute__((ext_vector_type(4)));
typedef int          i32x4 __attribute__((ext_vector_type(4)));
typedef int          i32x8 __attribute__((ext_vector_type(8)));

// Problem constants (B=4, S=1024 -> T=4096 tokens)
#define T_TOK 4096
#define DDIM  1024
#define ODIM  1024
#define NCAT  8

// GEMM tiling
#define BM 32            // tokens per workgroup tile
#define BN 128           // output cols per workgroup tile (wave strip = 16x32)
#define BK 32            // K block staged through LDS
#define LDA 36           // padded LDS stride for A (144B, 16B aligned, conflict-free)
#define LDB 136          // padded LDS stride for B (matches TDM pad: 128 + 8 dwords)

#if defined(__has_builtin)
#if __has_builtin(__builtin_amdgcn_tensor_load_to_lds)
#define HAVE_TDM 1
#endif
#endif

// ---------------- counting sort: group tokens by category ----------------

__global__ void cat_init_kernel(int* counts) {
    if (threadIdx.x < NCAT) counts[threadIdx.x] = 0;
}

__global__ void cat_hist_kernel(const int* __restrict__ cid, int* counts) {
    int t = blockIdx.x * blockDim.x + threadIdx.x;
    if (t < T_TOK) {
        int c = cid[t] & (NCAT - 1);
        atomicAdd(&counts[c], 1);
    }
}

__global__ void cat_scan_kernel(const int* __restrict__ counts,
                                int* offsets, int* cursor) {
    if (threadIdx.x == 0 && blockIdx.x == 0) {
        int acc = 0;
        for (int c = 0; c < NCAT; ++c) {
            offsets[c] = acc;
            cursor[c]  = acc;
            acc += counts[c];
        }
        offsets[NCAT] = acc;
    }
}

__global__ void cat_scatter_kernel(const int* __restrict__ cid,
                                   int* cursor, int* __restrict__ tokidx) {
    int t = blockIdx.x * blockDim.x + threadIdx.x;
    if (t < T_TOK) {
        int c = cid[t] & (NCAT - 1);
        int pos = atomicAdd(&cursor[c], 1);
        tokidx[pos] = t;
    }
}

// ---------------- TDM: async 2D weight-tile DMA into LDS ----------------
//
// D# per CDNA5 ISA:
//  group0: count=1 | lds_addr | global_addr | type=2
//  group1: data_size=2 (4B), pad_enable=1, pad_interval=6 (128 dwords),
//          pad_amount=7 (8 dwords)  -> LDS row stride = 136 floats,
//          tensor_dim0 = tile_dim0 = BN, tensor_dim1 = tile_dim1 = BK,
//          tensor_dim0_stride = ODIM (row-major W[cat])
//  groups 2/3: zero (2D tensor)

__device__ __forceinline__ void tdm_load_b(const float* gsrc, unsigned lds_byte_addr) {
#ifdef HAVE_TDM
    unsigned long long ga = (unsigned long long)(uintptr_t)gsrc;
    u32x4 g0;
    g0.x = 1u;                                   // count=1, user descriptor
    g0.y = lds_byte_addr;                        // lds_addr (bytes)
    g0.z = (unsigned)ga;                         // global_addr[31:0]
    g0.w = (unsigned)(ga >> 32) | (2u << 30);    // global_addr[56:32] | type=2

    i32x8 g1;
    g1[0] = (int)((2u << 16) | (1u << 20) | (6u << 22) | (7u << 25));
    g1[1] = (int)((unsigned)BN << 16);           // tensor_dim0[15:0] -> bits 63:48
    g1[2] = (int)((unsigned)BK << 16);           // tensor_dim1[15:0] -> bits 95:80
    g1[3] = (int)((unsigned)BN << 16);           // tile_dim0 -> bits 127:112
    g1[4] = (int)BK;                             // tile_dim1 = BK, tile_dim2 = 0
    g1[5] = (int)ODIM;                           // tensor_dim0_stride[31:0]
    g1[6] = 0;                                   // stride hi / dim1_stride lo
    g1[7] = 0;                                   // dim1_stride hi

    i32x4 z4 = {0, 0, 0, 0};
#if __clang_major__ >= 23
    i32x8 z8 = {0, 0, 0, 0, 0, 0, 0, 0};
    __builtin_amdgcn_tensor_load_to_lds(g0, g1, z4, z4, z8, 0);
#else
    __builtin_amdgcn_tensor_load_to_lds(g0, g1, z4, z4, 0);
#endif
#else
    (void)gsrc; (void)lds_byte_addr;
#endif
}

// ---------------- grouped gather-GEMM with WMMA f32 16x16x4 ----------------
//
// grid.x : ODIM / BN = 8 column tiles
// grid.y : NCAT * (T_TOK / BM) = 1024 row-tile slots (empty slots exit)
// block  : 256 threads = 8 waves; wave w owns a 16x32 strip:
//          mOff = (w&1)*16, nOff = (w>>1)*32, A fragment reused across 2 WMMAs.

__global__ __launch_bounds__(256)
void cat_gemm_kernel(const float* __restrict__ x,
                     const int*   __restrict__ tokidx,
                     const int*   __restrict__ offsets,
                     const float* __restrict__ weight,
                     const float* __restrict__ bias,
                     float*       __restrict__ out) {
    const int tilesPerCat = T_TOK / BM;
    const int cat  = blockIdx.y / tilesPerCat;
    const int tile = blockIdx.y % tilesPerCat;

    const int cbeg = offsets[cat];
    const int cnt  = offsets[cat + 1] - cbeg;
    const int rowBase = tile * BM;
    if (rowBase >= cnt) return;                 // uniform exit, EXEC stays all-1s

    const int oBase = blockIdx.x * BN;

    __shared__ float lA[2][BM * LDA];
    __shared__ float lB[2][BK * LDB];
    __shared__ int   lTok[BM];

    const int tid = threadIdx.x;

    // gather token ids for this row tile (clamp padding rows to a valid token)
    if (tid < BM) {
        int g = rowBase + tid;
        if (g >= cnt) g = cnt - 1;
        lTok[tid] = tokidx[cbeg + g];
    }
    __syncthreads();

    // cooperative A-load coordinates (32 rows x 32 k, one float4/thread)
    const int arow = tid >> 3;
    const int acol = (tid & 7) * 4;
    const int tokA = lTok[arow];

#ifndef HAVE_TDM
    // fallback B-load coordinates (32 rows x 128 n, four float4/thread)
    const int brow = tid >> 3;
    const int bcol = (tid & 7) * 16;
#endif

    const float* __restrict__ wc = weight + (size_t)cat * DDIM * ODIM;

    const int lane = tid & 31;
    const int wave = tid >> 5;
    const int mOff = (wave & 1) * 16;
    const int nOff = (wave >> 1) * 32;

    // WMMA fragment coordinates per ISA VGPR layouts
    const int kHi  = (lane >> 4) << 1;    // lanes 16-31 carry K+2,K+3
    const int mRow = mOff + (lane & 15);
    const int nCol = nOff + (lane & 15);

    v8f acc0 = {0.f, 0.f, 0.f, 0.f, 0.f, 0.f, 0.f, 0.f};
    v8f acc1 = {0.f, 0.f, 0.f, 0.f, 0.f, 0.f, 0.f, 0.f};

    // ---- prologue: start B DMA for block 0, prefetch A block 0 to regs ----
#ifdef HAVE_TDM
    if (wave == 0)
        tdm_load_b(wc + oBase, (unsigned)(uintptr_t)&lB[0][0]);
#else
    float4 bReg[4];
    #pragma unroll
    for (int j = 0; j < 4; ++j)
        bReg[j] = *(const float4*)(wc + (size_t)brow * ODIM + oBase + bcol + j * 4);
#endif
    float4 aReg = *(const float4*)(x + (size_t)tokA * DDIM + acol);

    for (int kb = 0, it = 0; kb < DDIM; kb += BK, ++it) {
        const int b = it & 1;

        // stage prefetched A regs (and fallback B regs) into buffer b
        *(float4*)&lA[b][arow * LDA + acol] = aReg;
#ifndef HAVE_TDM
        #pragma unroll
        for (int j = 0; j < 4; ++j)
            *(float4*)&lB[b][brow * LDB + bcol + j * 4] = bReg[j];
#endif

#ifdef HAVE_TDM
        __builtin_amdgcn_s_wait_tensorcnt(0);   // TDM for buffer b complete
#endif
        __syncthreads();                        // buffers b fully visible

        const int  kn   = kb + BK;
        const bool more = kn < DDIM;
        const int  kS   = more ? kn : 0;

        // start next B DMA into buffer b^1 (safe: last readers of b^1 passed
        // the barrier above) and prefetch next A block into registers
#ifdef HAVE_TDM
        if (more && wave == 0)
            tdm_load_b(wc + (size_t)kn * ODIM + oBase,
                       (unsigned)(uintptr_t)&lB[b ^ 1][0]);
#else
        if (more) {
            #pragma unroll
            for (int j = 0; j < 4; ++j)
                bReg[j] = *(const float4*)(wc + (size_t)(kn + brow) * ODIM + oBase + bcol + j * 4);
        }
#endif
        aReg = *(const float4*)(x + (size_t)tokA * DDIM + kS + acol);

        const float* __restrict__ la = &lA[b][0];
        const float* __restrict__ lb = &lB[b][0];

        #pragma unroll
        for (int kk = 0; kk < BK; kk += 4) {
            v2f a, b0, b1;
            // A 16x4 f32: lane m = lane&15, VGPRs = K+kHi, K+kHi+1
            a.x  = la[mRow * LDA + kk + kHi];
            a.y  = la[mRow * LDA + kk + kHi + 1];
            // B 4x16 f32: lane n = lane&15, rows K+kHi, K+kHi+1
            b0.x = lb[(kk + kHi)     * LDB + nCol];
            b0.y = lb[(kk + kHi + 1) * LDB + nCol];
            b1.x = lb[(kk + kHi)     * LDB + nCol + 16];
            b1.y = lb[(kk + kHi + 1) * LDB + nCol + 16];
            acc0 = __builtin_amdgcn_wmma_f32_16x16x4_f32(
                false, a, false, b0, (short)0, acc0, false, false);
            acc1 = __builtin_amdgcn_wmma_f32_16x16x4_f32(
                false, a, false, b1, (short)0, acc1, false, false);
        }
        // no trailing barrier: both A and B are double-buffered
    }

    // epilogue: C/D layout -> lane l, VGPR r holds (M = r + 8*(l>=16), N = l&15)
    const int n0 = oBase + nCol;
    const float bv0 = bias[cat * ODIM + n0];
    const float bv1 = bias[cat * ODIM + n0 + 16];
    #pragma unroll
    for (int r = 0; r < 8; ++r) {
        int m = r + ((lane >> 4) << 3);
        int g = rowBase + mOff + m;
        if (g < cnt) {
            int token = lTok[mOff + m];
            out[(size_t)token * ODIM + n0]      = acc0[r] + bv0;
            out[(size_t)token * ODIM + n0 + 16] = acc1[r] + bv1;
        }
    }
}

// ---------------- launch ----------------

extern "C" void kernel_launch(void* const* d_in, const int* in_sizes, int n_in,
                              void* d_out, int out_size, void* d_ws, size_t ws_size,
                              hipStream_t stream) {
    const float* x    = (const float*)d_in[0];
    const int*   cid  = (const int*)d_in[1];   // JAX default x64-off -> int32
    const float* w    = (const float*)d_in[2];
    const float* bias = (const float*)d_in[3];
    float*       out  = (float*)d_out;

    int* ws      = (int*)d_ws;
    int* counts  = ws;        // [8]
    int* offsets = ws + 8;    // [9]
    int* cursor  = ws + 17;   // [8]
    int* tokidx  = ws + 32;   // [T_TOK]

    cat_init_kernel<<<1, 32, 0, stream>>>(counts);
    cat_hist_kernel<<<T_TOK / 256, 256, 0, stream>>>(cid, counts);
    cat_scan_kernel<<<1, 1, 0, stream>>>(counts, offsets, cursor);
    cat_scatter_kernel<<<T_TOK / 256, 256, 0, stream>>>(cid, cursor, tokidx);

    dim3 grid(ODIM / BN, NCAT * (T_TOK / BM));
    cat_gemm_kernel<<<grid, 256, 0, stream>>>(x, tokidx, offsets, w, bias, out);
}